// LossMatch_5033701671390
// MI455X (gfx1250) — compile-verified
//
#include <hip/hip_runtime.h>

typedef __attribute__((ext_vector_type(2))) float v2f;
typedef __attribute__((ext_vector_type(4))) float v4f;
typedef __attribute__((ext_vector_type(8))) float v8f;

constexpr int       B_   = 256;
constexpr int       K_   = 32;
constexpr int       D_   = 2048;
constexpr long long N_   = 100000;
constexpr int       KCH  = 4;          // k-chunks per batch element
constexpr int       KPER = K_ / KCH;   // 8 k's per block
constexpr int       TPB  = 256;        // 8 waves (wave32)
constexpr int       UNR  = 8;          // loads kept in flight per wave

// Each block handles one (b, k-chunk): KPER*D = 16384 contiguous floats of e.
// 256 threads * float4 = 1024 floats/iter -> 16 iters. d = (it*1024 + t*4) % 2048,
// so each thread only ever touches d0 = 4t (even iters) and 1024 + 4t (odd iters):
// 8 target values, gathered once into registers.
__global__ void __launch_bounds__(TPB)
loss_match_partial(const float* __restrict__ e,
                   const float* __restrict__ W,
                   const long long* __restrict__ idx,
                   float* __restrict__ partials) {
    const int t  = threadIdx.x;
    const int b  = blockIdx.x;
    const int kc = blockIdx.y;

    const long long col = idx[b];
    const int d0 = t * 4;

    // One-time scattered gather of this thread's 8 target values (stride-N column
    // of W). These lines are reused across the KCH chunks of the same b -> let
    // them live in GL2 (regular temporal loads).
    float tg[8];
#pragma unroll
    for (int j = 0; j < 4; ++j) {
        tg[j]     = W[(long long)(d0 + j)        * N_ + col];
        tg[4 + j] = W[(long long)(d0 + 1024 + j) * N_ + col];
    }

    const float* eb = e + ((long long)b * K_ + (long long)kc * KPER) * (long long)D_;

    v8f acc = {};                 // 16x16 f32 WMMA accumulator (8 VGPRs)
    v2f ones; ones[0] = 1.0f; ones[1] = 1.0f;   // B = all-ones 4x16

#pragma unroll UNR
    for (int it = 0; it < 16; ++it) {
        // e is streamed exactly once: non-temporal b128 load, don't pollute GL2.
        const v4f ev = __builtin_nontemporal_load((const v4f*)(eb + it * 1024 + d0));
        // prefetch a full unrolled iteration ahead (UNR * 4 KiB per thread-line)
        __builtin_prefetch(eb + (it + UNR) * 1024 + d0, 0, 0);
        const float* tp = tg + ((it & 1) ? 4 : 0);
        v2f a01, a23;
        a01[0] = fabsf(ev[0] - tp[0]);
        a01[1] = fabsf(ev[1] - tp[1]);
        a23[0] = fabsf(ev[2] - tp[2]);
        a23[1] = fabsf(ev[3] - tp[3]);
        // D = A x ones + C : folds 64 abs-diffs per wave per instruction.
        acc = __builtin_amdgcn_wmma_f32_16x16x4_f32(false, a01, false, ones,
                                                    (short)0, acc, false, false);
        acc = __builtin_amdgcn_wmma_f32_16x16x4_f32(false, a23, false, ones,
                                                    (short)0, acc, false, false);
    }

    // Each lane sums its 8 accumulator entries; ones-B replicated the sum over
    // 16 columns, so the block total is 16x the true sum (divided out at the end).
    float s = acc[0] + acc[1] + acc[2] + acc[3] + acc[4] + acc[5] + acc[6] + acc[7];

    // wave32 butterfly reduction
#pragma unroll
    for (int m = 16; m >= 1; m >>= 1)
        s += __shfl_xor(s, m, 32);

    __shared__ float wsum[TPB / 32];
    if ((t & 31) == 0) wsum[t >> 5] = s;
    __syncthreads();

    if (t == 0) {
        float tot = 0.0f;
#pragma unroll
        for (int w = 0; w < TPB / 32; ++w) tot += wsum[w];
        partials[blockIdx.y * gridDim.x + blockIdx.x] = tot;  // overwrite: no init needed
    }
}

__global__ void __launch_bounds__(TPB)
loss_match_final(const float* __restrict__ partials, float* __restrict__ out) {
    const int t = threadIdx.x;
    float s = 0.0f;
#pragma unroll
    for (int i = t; i < B_ * KCH; i += TPB) s += partials[i];

#pragma unroll
    for (int m = 16; m >= 1; m >>= 1)
        s += __shfl_xor(s, m, 32);

    __shared__ float wsum[TPB / 32];
    if ((t & 31) == 0) wsum[t >> 5] = s;
    __syncthreads();

    if (t == 0) {
        float tot = 0.0f;
#pragma unroll
        for (int w = 0; w < TPB / 32; ++w) tot += wsum[w];
        // mean over B*K*D, WMMA ones-column duplication factor 16, weight 80.
        out[0] = tot * (80.0f / (16.0f * (float)B_ * (float)K_ * (float)D_));
    }
}

extern "C" void kernel_launch(void* const* d_in, const int* in_sizes, int n_in,
                              void* d_out, int out_size, void* d_ws, size_t ws_size,
                              hipStream_t stream) {
    const float*     e   = (const float*)d_in[0];       // [B,K,D,1] f32
    const float*     W   = (const float*)d_in[1];       // [D,N]     f32
    const long long* idx = (const long long*)d_in[2];   // [B]       int64
    float* partials = (float*)d_ws;                     // B*KCH = 1024 floats
    float* out      = (float*)d_out;

    dim3 grid(B_, KCH);
    hipLaunchKernelGGL(loss_match_partial, grid, dim3(TPB), 0, stream,
                       e, W, idx, partials);
    hipLaunchKernelGGL(loss_match_final, dim3(1), dim3(TPB), 0, stream,
                       partials, out);
}